// TokenMerger_37778532336201
// MI455X (gfx1250) — compile-verified
//
#include <hip/hip_runtime.h>

typedef __attribute__((ext_vector_type(2))) float v2f;
typedef __attribute__((ext_vector_type(8))) float v8f;

#define BB 128
#define NN 256
#define DD 256
#define NP 257            // padded LDS row stride: 257 % 64 == 1 -> conflict-free rows AND columns
#define THRESH 0.2f

// ---------------------------------------------------------------------------
// Fused per-batch megakernel: coalesced wave-cooperative norms -> WMMA Gram
// matrix (upper triangle only, mirrored store) into 320KB WGP LDS -> greedy
// merge loop (incremental row-max argmax, exact early exit) -> merge_map out.
// One workgroup (256 threads = 8 wave32) per batch.
// ---------------------------------------------------------------------------
__global__ void __launch_bounds__(256)
tokenmerge_fused_kernel(const float* __restrict__ slots,
                        int* __restrict__ mt_out,
                        float* __restrict__ mm_out)
{
    __shared__ float s_sim[NN * NP];   // 263,168 B
    __shared__ float s_inv[NN];
    __shared__ float s_rowmax[NN];
    __shared__ int   s_rowarg[NN];
    __shared__ int   s_dead[NN];
    __shared__ int   s_mt[NN];
    __shared__ float s_redV[256];
    __shared__ int   s_redI[256];

    const int b    = blockIdx.x;
    const int t    = threadIdx.x;       // 0..255
    const int wave = t >> 5;            // 0..7 (wave32)
    const int lane = t & 31;

    const float* base = slots + (size_t)b * NN * DD;

    // ---- Phase 0: per-row inverse norms, wave-cooperative (coalesced) ----
    {
        const int r0 = wave * 32;
        for (int i = 0; i < 32; ++i) {
            const int row = r0 + i;
            const float* rp = base + (size_t)row * DD;
            float ss = 0.0f;
            for (int k = lane; k < DD; k += 32) { float v = rp[k]; ss += v * v; }
            #pragma unroll
            for (int m = 16; m > 0; m >>= 1) ss += __shfl_xor(ss, m, 32);
            if (lane == 0) s_inv[row] = 1.0f / fmaxf(sqrtf(ss), 1e-12f);
        }
    }
    s_mt[t]   = t;
    s_dead[t] = 0;
    __syncthreads();

    // ---- Phase 1: sim = sn*sn^T - 2I via V_WMMA_F32_16X16X4_F32 into LDS ----
    // Symmetric: compute only 136 upper-triangle 16x16 tiles, mirror the store.
    {
        const int lrow = lane & 15;
        const int koff = (lane < 16) ? 0 : 2;   // fp32 A/B K-split across lane halves
        for (int p = wave; p < 136; p += 8) {
            // map linear p -> (ti, tj) with ti <= tj
            int ti = 0, rem = p;
            while (rem >= 16 - ti) { rem -= 16 - ti; ++ti; }
            const int tj = ti + rem;

            const int rowA = ti * 16 + lrow;
            const int rowB = tj * 16 + lrow;
            const float ia = s_inv[rowA];
            const float ib = s_inv[rowB];
            const float2* pA = (const float2*)(base + (size_t)rowA * DD);
            const float2* pB = (const float2*)(base + (size_t)rowB * DD);
            v8f acc = {0.f, 0.f, 0.f, 0.f, 0.f, 0.f, 0.f, 0.f};
            for (int k = 0; k < DD; k += 4) {
                float2 a2 = pA[(k + koff) >> 1];
                float2 b2 = pB[(k + koff) >> 1];
                v2f av; av[0] = a2.x * ia; av[1] = a2.y * ia;
                v2f bv; bv[0] = b2.x * ib; bv[1] = b2.y * ib;
                acc = __builtin_amdgcn_wmma_f32_16x16x4_f32(
                          false, av, false, bv, (short)0, acc, false, false);
            }
            // C/D layout: VGPR g -> M = g (lanes 0-15) / g+8 (lanes 16-31), N = lane%16
            const int crow0 = (lane < 16) ? 0 : 8;
            const int gj    = tj * 16 + (lane & 15);
            #pragma unroll
            for (int g = 0; g < 8; ++g) {
                const int gi = ti * 16 + crow0 + g;
                float v = acc[g];
                if (gi == gj) v -= 2.0f;              // sim -= 2*I
                s_sim[gi * NP + gj] = v;
                if (ti != tj) s_sim[gj * NP + gi] = v; // symmetric mirror (conflict-free)
            }
        }
    }
    __syncthreads();

    // ---- Phase 2: initial per-row maxima (first-occurrence tie-break) ----
    {
        const float* row = &s_sim[t * NP];
        float best = -3.0f; int barg = 0;
        for (int j = 0; j < NN; ++j) {
            float v = row[j];
            if (v > best) { best = v; barg = j; }
        }
        s_rowmax[t] = best;
        s_rowarg[t] = barg;
    }
    __syncthreads();

    // ---- Phase 3: greedy merge loop ----
    for (int iter = 0; iter < NN; ++iter) {
        // global argmax over row maxima; ties -> smaller row (== smaller flat idx)
        s_redV[t] = s_dead[t] ? -3.0f : s_rowmax[t];
        s_redI[t] = t;
        __syncthreads();
        for (int s = 128; s > 0; s >>= 1) {
            if (t < s) {
                float v1 = s_redV[t],     v2 = s_redV[t + s];
                int   i1 = s_redI[t],     i2 = s_redI[t + s];
                if (v2 > v1 || (v2 == v1 && i2 < i1)) { s_redV[t] = v2; s_redI[t] = i2; }
            }
            __syncthreads();
        }
        const float mx = s_redV[0];
        const int   r  = s_redI[0];
        if (mx <= THRESH) break;                 // exact early exit (uniform)

        const int c   = s_rowarg[r];
        const int src = (r > c) ? r : c;
        const int tgt = (r > c) ? c : r;
        if (t == 0) { s_mt[src] = tgt; s_dead[src] = 1; }
        __syncthreads();

        // only rows whose current argmax column just died need a rescan
        if (!s_dead[t] && s_rowarg[t] == src) {
            const float* row = &s_sim[t * NP];
            float best = -3.0f; int barg = 0;
            for (int j = 0; j < NN; ++j) {
                if (!s_dead[j]) {
                    float v = row[j];
                    if (v > best) { best = v; barg = j; }
                }
            }
            s_rowmax[t] = best;
            s_rowarg[t] = barg;
        }
        __syncthreads();
    }

    mt_out[b * NN + t] = s_mt[t];
    mm_out[b * NN + t] = (float)s_mt[t];
}

// ---------------------------------------------------------------------------
// Scatter-merge: merged[b,n,:] = sum_{s: mt[s]==n} slots[b,s,:] / max(count,1)
// One block per (n, b); thread d owns feature d (coalesced loads per match).
// ---------------------------------------------------------------------------
__global__ void __launch_bounds__(256)
tokenmerge_scatter_kernel(const float* __restrict__ slots,
                          const int* __restrict__ mt,
                          float* __restrict__ out)
{
    const int b = blockIdx.y;
    const int n = blockIdx.x;
    const int d = threadIdx.x;

    __shared__ int s_mt[NN];
    s_mt[d] = mt[b * NN + d];
    __syncthreads();

    float acc = 0.0f;
    int cnt = 0;
    const float* base = slots + (size_t)b * NN * DD;
    for (int s = 0; s < NN; ++s) {
        if (s_mt[s] == n) { acc += base[(size_t)s * DD + d]; ++cnt; }
    }
    out[((size_t)(b * NN + n)) * DD + d] = acc / fmaxf((float)cnt, 1.0f);
}

extern "C" void kernel_launch(void* const* d_in, const int* in_sizes, int n_in,
                              void* d_out, int out_size, void* d_ws, size_t ws_size,
                              hipStream_t stream)
{
    (void)in_sizes; (void)n_in; (void)out_size; (void)ws_size;

    const float* slots = (const float*)d_in[0];
    float* out_f       = (float*)d_out;
    float* merged      = out_f;                            // B*N*D floats
    float* merge_map_f = out_f + (size_t)BB * NN * DD;     // B*N floats (int values)
    int*   mt_ws       = (int*)d_ws;                       // B*N ints of scratch

    tokenmerge_fused_kernel<<<dim3(BB), dim3(256), 0, stream>>>(slots, mt_ws, merge_map_f);
    tokenmerge_scatter_kernel<<<dim3(NN, BB), dim3(256), 0, stream>>>(slots, mt_ws, merged);
}